// DualSpacesIntervention_30502857736485
// MI455X (gfx1250) — compile-verified
//
#include <hip/hip_runtime.h>
#include <stdint.h>

typedef __attribute__((ext_vector_type(2))) float v2f;
typedef __attribute__((ext_vector_type(8))) float v8f;

#define DIM        4096
#define KLR        16          // low-rank k
#define RLR        32          // 2k
#define TILE_M     16
#define ROW_STRIDE 4100        // padded LDS row stride (floats): conflict-free frags
#define T_STRIDE   18          // padded stride for the small t tile (even -> 8B aligned)

// ---------------------------------------------------------------------------
// prep: mid[j][i] = sum_p B[p][j] * e[p] * aa[i][p]   (16x16)
//       bb2[d][j] = sum_i bb[d][i] * mid[j][i]        (4096x16)
// After this, out = base + (base @ A^T) @ bb2^T.
// ---------------------------------------------------------------------------
__global__ void __launch_bounds__(256)
prep_kernel(const float* __restrict__ lora_B,   // [32,16]
            const float* __restrict__ lora_E,   // [32]
            const float* __restrict__ lora_aa,  // [16,32]
            const float* __restrict__ lora_bb,  // [4096,16]
            float* __restrict__ bb2)            // [4096,16]
{
    __shared__ float mid[KLR][KLR];
    const int tid = threadIdx.x;
    {
        const int j = tid >> 4, i = tid & 15;
        float s = 0.f;
#pragma unroll
        for (int p = 0; p < RLR; ++p)
            s += lora_B[p * KLR + j] * lora_E[p] * lora_aa[i * RLR + p];
        mid[j][i] = s;
    }
    __syncthreads();

    const int d = blockIdx.x * 256 + tid;
    float row[KLR];
#pragma unroll
    for (int i = 0; i < KLR; ++i) row[i] = lora_bb[(size_t)d * KLR + i];
#pragma unroll
    for (int j = 0; j < KLR; ++j) {
        float s = 0.f;
#pragma unroll
        for (int i = 0; i < KLR; ++i) s += row[i] * mid[j][i];
        bb2[(size_t)d * KLR + j] = s;
    }
}

// ---------------------------------------------------------------------------
// fused: per 16-row tile:
//   1) async-DMA base tile into LDS (single HBM read of base, no VGPR bounce)
//   2) t = base_tile @ A^T          (wmma f32 16x16x4, K split over 4 waves)
//   3) out = t @ bb2^T + base_tile  (base folded into WMMA C-operand)
// ---------------------------------------------------------------------------
__global__ void __launch_bounds__(128)
fused_kernel(const float* __restrict__ base,    // [16384,4096]
             const float* __restrict__ lora_A,  // [16,4096]
             const float* __restrict__ bb2,     // [4096,16]
             float* __restrict__ out)           // [16384,4096]
{
    extern __shared__ float lds[];
    float* baseT = lds;                               // 16 * ROW_STRIDE
    float* red   = baseT + TILE_M * ROW_STRIDE;       // 4 * 16 * T_STRIDE
    float* tT    = red + 4 * 16 * T_STRIDE;           // 16 * T_STRIDE

    const int tid  = threadIdx.x;
    const int lane = tid & 31;
    const int wave = tid >> 5;
    const int m    = lane & 15;       // row (A frag) or col (B/C frag) index
    const int hi   = lane >> 4;
    const int koff = hi * 2;          // K sub-offset per ISA A/B layout
    const int row0 = blockIdx.x * TILE_M;

    // ---- Phase 1: async global->LDS DMA of the base tile (b128/lane) ----
    {
        // LDS byte offset = low 32 bits of the generic pointer (ISA flat->LDS rule)
        const uint32_t ldsT = (uint32_t)(uintptr_t)baseT;
        for (int r = 0; r < TILE_M; ++r) {
            const char* srow = (const char*)(base + (size_t)(row0 + r) * DIM);
            const uint32_t drow = ldsT + (uint32_t)(r * ROW_STRIDE * 4);
#pragma unroll
            for (int i = 0; i < 8; ++i) {
                const uint32_t dst = drow + (uint32_t)((tid + i * 128) * 16);
                const char* src = srow + (size_t)(tid + i * 128) * 16;
                asm volatile("global_load_async_to_lds_b128 %0, %1, off"
                             :: "v"(dst), "v"(src)
                             : "memory");
            }
        }
        asm volatile("s_wait_asynccnt 0x0" ::: "memory");
    }
    __syncthreads();

    // ---- Phase 2: GEMM1 partial  t_w = base[:, k0:k0+1024] @ A[:, k0:k0+1024]^T ----
    v8f acc = {};
    const float* aFrag = baseT + m * ROW_STRIDE + wave * 1024 + koff;
    const float* bFrag = lora_A + (size_t)m * DIM + wave * 1024 + koff;
#pragma unroll 4
    for (int it = 0; it < 256; ++it) {
        v2f a = *reinterpret_cast<const v2f*>(aFrag + it * 4);
        v2f b = *reinterpret_cast<const v2f*>(bFrag + it * 4);
        acc = __builtin_amdgcn_wmma_f32_16x16x4_f32(false, a, false, b,
                                                    (short)0, acc, false, false);
    }
    // spill per-wave partial C (M = g + 8*hi, N = m) and reduce across waves
#pragma unroll
    for (int g = 0; g < 8; ++g)
        red[wave * (16 * T_STRIDE) + (g + 8 * hi) * T_STRIDE + m] = acc[g];
    __syncthreads();

#pragma unroll
    for (int q = 0; q < 2; ++q) {
        const int idx = tid * 2 + q;
        const int mm = idx >> 4, nn = idx & 15;
        float s = red[0 * (16 * T_STRIDE) + mm * T_STRIDE + nn]
                + red[1 * (16 * T_STRIDE) + mm * T_STRIDE + nn]
                + red[2 * (16 * T_STRIDE) + mm * T_STRIDE + nn]
                + red[3 * (16 * T_STRIDE) + mm * T_STRIDE + nn];
        tT[mm * T_STRIDE + nn] = s;
    }
    __syncthreads();

    // ---- Phase 3: out[:, c0:c0+16] = t @ bb2[c0:c0+16, :]^T + base tile ----
#pragma unroll 1
    for (int b = 0; b < 64; ++b) {
        const int c0 = wave * 1024 + b * 16;
        v8f c;
#pragma unroll
        for (int g = 0; g < 8; ++g)                    // C init = base (fused add)
            c[g] = baseT[(g + 8 * hi) * ROW_STRIDE + c0 + m];
#pragma unroll
        for (int kk = 0; kk < KLR; kk += 4) {
            v2f a = *reinterpret_cast<const v2f*>(tT + m * T_STRIDE + kk + koff);
            v2f bf = *reinterpret_cast<const v2f*>(bb2 + (size_t)(c0 + m) * KLR + kk + koff);
            c = __builtin_amdgcn_wmma_f32_16x16x4_f32(false, a, false, bf,
                                                      (short)0, c, false, false);
        }
#pragma unroll
        for (int g = 0; g < 8; ++g)
            out[(size_t)(row0 + g + 8 * hi) * DIM + c0 + m] = c[g];
    }
}

extern "C" void kernel_launch(void* const* d_in, const int* in_sizes, int n_in,
                              void* d_out, int out_size, void* d_ws, size_t ws_size,
                              hipStream_t stream) {
    const float* base    = (const float*)d_in[0];   // [4,4096,4096]
    const float* lora_A  = (const float*)d_in[1];   // [16,4096]
    const float* lora_E  = (const float*)d_in[2];   // [32,1]
    const float* lora_B  = (const float*)d_in[3];   // [32,16]
    const float* lora_aa = (const float*)d_in[4];   // [16,32]
    const float* lora_bb = (const float*)d_in[5];   // [4096,16]
    float* out = (float*)d_out;
    float* bb2 = (float*)d_ws;                      // 4096*16 floats = 256 KB scratch

    prep_kernel<<<DIM / 256, 256, 0, stream>>>(lora_B, lora_E, lora_aa, lora_bb, bb2);

    const size_t lds_bytes =
        (size_t)(TILE_M * ROW_STRIDE + 4 * 16 * T_STRIDE + 16 * T_STRIDE) * sizeof(float);
    const int n_tiles = (4 * 4096) / TILE_M;        // 16384 rows / 16 = 1024 tiles
    fused_kernel<<<n_tiles, 128, lds_bytes, stream>>>(base, lora_A, bb2, out);
}